// SeesawLoss_42253888258851
// MI455X (gfx1250) — compile-verified
//
#include <hip/hip_runtime.h>
#include <cstdint>

#define NUM_CLASSES 32000
#define BATCH       8192
#define LOG2E_F     1.4426950408889634f
#define LN2_F       0.6931471805599453f

typedef __attribute__((ext_vector_type(2))) float v2f;
typedef __attribute__((ext_vector_type(8))) float v8f;

// ---------------------------------------------------------------------------
// K0: zero the class-count histogram (workspace is poisoned, must re-zero
// every launch for determinism).
// ---------------------------------------------------------------------------
__global__ void seesaw_zero_counts(uint32_t* __restrict__ counts) {
    int i = blockIdx.x * blockDim.x + threadIdx.x;
    if (i < NUM_CLASSES) counts[i] = 0u;
}

// ---------------------------------------------------------------------------
// K1: histogram of targets (global_atomic_add_u32; integer => deterministic).
// ---------------------------------------------------------------------------
__global__ void seesaw_histogram(const int* __restrict__ targets,
                                 uint32_t* __restrict__ counts) {
    int i = blockIdx.x * blockDim.x + threadIdx.x;
    if (i < BATCH) atomicAdd(&counts[targets[i]], 1u);
}

// ---------------------------------------------------------------------------
// K2: one block (8 wave32s) per row. Single-pass online log-sum-exp in the
// base-2 domain (native v_exp_f32), B128 loads + global_prefetch_b8.
// Writes  w_unnorm(t_b) * (x[b,t_b] - lse_b)  to row_contrib[b].
// ---------------------------------------------------------------------------
__device__ __forceinline__ void lse_update(float x, float& m2, float& s) {
    float y = x * LOG2E_F;          // work in log2 domain
    float d = y - m2;
    if (d > 0.0f) {                 // rare after warmup: rescale accumulator
        s = s * exp2f(-d) + 1.0f;
        m2 = y;
    } else {
        s += exp2f(d);              // common path: one v_exp_f32
    }
}

__global__ void __launch_bounds__(256) seesaw_row_kernel(
        const float* __restrict__ logits,
        const int* __restrict__ targets,
        const uint32_t* __restrict__ counts,
        float* __restrict__ row_contrib) {
    const int b   = blockIdx.x;
    const int tid = threadIdx.x;
    const float4* rowv = (const float4*)(logits + (size_t)b * NUM_CLASSES);
    const int NV = NUM_CLASSES / 4;   // 8000 float4 per row

    float m2 = -INFINITY, s = 0.0f;
    for (int i = tid; i < NV; i += 256) {
        __builtin_prefetch((const void*)(rowv + i + 1024), 0, 3); // ~16KB ahead
        float4 v = rowv[i];
        lse_update(v.x, m2, s);
        lse_update(v.y, m2, s);
        lse_update(v.z, m2, s);
        lse_update(v.w, m2, s);
    }

    // wave32 (m,s) combine
    for (int off = 16; off > 0; off >>= 1) {
        float mo = __shfl_xor(m2, off, 32);
        float so = __shfl_xor(s,  off, 32);
        float mn = fmaxf(m2, mo);
        s  = s * exp2f(m2 - mn) + so * exp2f(mo - mn);
        m2 = mn;
    }

    __shared__ float sm[8], ss[8];
    if ((tid & 31) == 0) { sm[tid >> 5] = m2; ss[tid >> 5] = s; }
    __syncthreads();

    if (tid == 0) {
        float M = sm[0], S = ss[0];
        #pragma unroll
        for (int w = 1; w < 8; ++w) {
            float mo = sm[w], so = ss[w];
            float mn = fmaxf(M, mo);
            S = S * exp2f(M - mn) + so * exp2f(mo - mn);
            M = mn;
        }
        float lse = (M + log2f(S)) * LN2_F;           // back to natural log
        int   t   = targets[b];
        float xt  = logits[(size_t)b * NUM_CLASSES + t];
        float cnt = (float)counts[t];
        float wun = exp2f(-0.8f * log2f(cnt + 0.01f)); // (cum+eps)^-0.8
        row_contrib[b] = wun * (xt - lse);
    }
}

// ---------------------------------------------------------------------------
// K3: (a) min over counts -> mit normalizer, (b) deterministic exact-f32 sum
// of the 8192 row contributions using V_WMMA_F32_16X16X4_F32 with an all-ones
// B matrix: D[m,n] = sum_k A[m,k] + C[m,n].  One wave folds 64 f32 / issue.
// ---------------------------------------------------------------------------
__global__ void __launch_bounds__(256) seesaw_finalize(
        const uint32_t* __restrict__ counts,
        const float* __restrict__ row_contrib,
        float* __restrict__ out) {
    __shared__ uint32_t redmin[256];
    const int tid = threadIdx.x;

    uint32_t mn = 0xFFFFFFFFu;
    for (int i = tid; i < NUM_CLASSES; i += 256) {
        uint32_t c = counts[i];
        mn = (c < mn) ? c : mn;
    }
    redmin[tid] = mn;
    __syncthreads();
    for (int off = 128; off > 0; off >>= 1) {
        if (tid < off) {
            uint32_t o = redmin[tid + off];
            if (o < redmin[tid]) redmin[tid] = o;
        }
        __syncthreads();
    }
    uint32_t min_cnt = redmin[0];

    if (tid < 32) {                    // wave 0, full EXEC (WMMA requirement)
        const int lane = tid;
        // A-matrix 16x4 f32 layout: lanes 0-15 hold (M=lane, K=0|1),
        // lanes 16-31 hold (M=lane-16, K=2|3). Any bijection works for a sum.
        const int base_off = 4 * (lane & 15) + ((lane >> 4) << 1);
        v2f bones; bones[0] = 1.0f; bones[1] = 1.0f;   // all-ones B (4x16)
        v8f acc = {};
        #pragma unroll 4
        for (int it = 0; it < BATCH / 64; ++it) {      // 128 iterations
            v2f a = *(const v2f*)(row_contrib + it * 64 + base_off);
            acc = __builtin_amdgcn_wmma_f32_16x16x4_f32(
                false, a, false, bones, (short)0, acc, false, false);
        }
        // acc[v] = per-M partial sums, replicated across columns. Column 0 of
        // D lives in lane 0 (M=0..7) and lane 16 (M=8..15).
        float lane_sum = acc[0] + acc[1] + acc[2] + acc[3] +
                         acc[4] + acc[5] + acc[6] + acc[7];
        float total = __shfl(lane_sum, 0, 32) + __shfl(lane_sum, 16, 32);
        if (lane == 0) {
            float mit_max = exp2f(-0.8f * log2f((float)min_cnt + 0.01f));
            out[0] = -(total / mit_max) / (float)BATCH;
        }
    }
}

// ---------------------------------------------------------------------------
extern "C" void kernel_launch(void* const* d_in, const int* in_sizes, int n_in,
                              void* d_out, int out_size, void* d_ws, size_t ws_size,
                              hipStream_t stream) {
    const float* logits  = (const float*)d_in[0];
    const int*   targets = (const int*)d_in[1];

    uint32_t* counts      = (uint32_t*)d_ws;                             // 128000 B
    float*    row_contrib = (float*)((char*)d_ws + NUM_CLASSES * 4);     //  32768 B
    float*    out         = (float*)d_out;

    seesaw_zero_counts<<<(NUM_CLASSES + 255) / 256, 256, 0, stream>>>(counts);
    seesaw_histogram<<<(BATCH + 255) / 256, 256, 0, stream>>>(targets, counts);
    seesaw_row_kernel<<<BATCH, 256, 0, stream>>>(logits, targets, counts, row_contrib);
    seesaw_finalize<<<1, 256, 0, stream>>>(counts, row_contrib, out);
}